// HierarchicalLoss_10342281248967
// MI455X (gfx1250) — compile-verified
//
#include <hip/hip_runtime.h>
#include <math.h>

// HierarchicalLoss for MI455X (gfx1250): wave32, fp32 WMMA, async global->LDS.
//
// loss = 0.5*CE + 0.5*mean_i[ 1 + dot(probs_i, H[t_i]) ]
// Memory-bound: ~256MB logits + ~64MB H (L2-resident) => ~14us floor @ 23.3TB/s.
// Per 16-row tile: phase 1 computes row maxima (coalesced streams, warms L2);
// phase 2 stages 16x32 chunks of logits and target-gathered H rows into LDS via
// GLOBAL_LOAD_ASYNC_TO_LDS_B128 (double-buffered, ASYNCcnt-synced, no WG
// barriers), applies v_exp_f32 on the A fragment, accumulates the per-row dots
// on the matrix pipe with V_WMMA_F32_16X16X4_F32 (diagonal of A x B).

typedef __attribute__((ext_vector_type(2))) float v2f;
typedef __attribute__((ext_vector_type(8))) float v8f;

#define ALPHA_  0.5f
#define ROWS    16            // tile rows per workgroup = WMMA M = N
#define NWAVES  4
#define BLOCK   (NWAVES * 32)
#define CHUNK   32            // columns staged per chunk per wave
#define LSTRIDE 36            // padded LDS row stride: 36i mod 64 permutes the
                              // multiples of 4 -> b64 fragment reads conflict-free

__global__ __launch_bounds__(BLOCK) void hier_loss_kernel(
    const float* __restrict__ logits,
    const int*   __restrict__ targets,
    const float* __restrict__ H,
    float*       __restrict__ partial,
    int Bn, int C)
{
    __shared__ float sLog[NWAVES][2][ROWS * LSTRIDE];   // double-buffered
    __shared__ float sHm [NWAVES][2][ROWS * LSTRIDE];
    __shared__ float sM[ROWS], sZ[ROWS], sDot[ROWS], sXt[ROWS];
    __shared__ int   sTgt[ROWS];

    const int tid  = threadIdx.x;
    const int lane = tid & 31;
    const int wave = tid >> 5;
    const int row0 = blockIdx.x * ROWS;

    // ---- setup: targets, target logit (for CE), zero accumulators ----
    if (tid < ROWS) {
        int t = targets[row0 + tid];
        sTgt[tid] = t;
        sXt[tid]  = logits[(size_t)(row0 + tid) * C + t];
        sZ[tid]   = 0.0f;
        sDot[tid] = 0.0f;
    }

    // ---- phase 1: per-row max (wave w owns rows 4w..4w+3), coalesced b128 ----
    #pragma unroll
    for (int rr = 0; rr < 4; ++rr) {
        const int r = wave * 4 + rr;
        const float4* rp = (const float4*)(logits + (size_t)(row0 + r) * C);
        float mx = -INFINITY;
        for (int k = lane; k < (C >> 2); k += 32) {
            float4 v = rp[k];
            mx = fmaxf(mx, fmaxf(fmaxf(v.x, v.y), fmaxf(v.z, v.w)));
        }
        #pragma unroll
        for (int off = 16; off; off >>= 1)
            mx = fmaxf(mx, __shfl_xor(mx, off, 32));
        if (lane == 0) sM[r] = mx;
    }
    __syncthreads();

    // ---- per-lane precompute ----
    // WMMA fragment mapping: lane L = h*16 + i holds tile[i][k0+2h], tile[i][k0+2h+1]
    const int   fi   = lane & 15;
    const int   fh   = lane >> 4;
    const int   aoff = fi * LSTRIDE + 2 * fh;
    const float mrow = sM[fi];

    // staging mapping: 128 float4 slots (16 rows x 8 segs), 4 per lane.
    // goff*: 32-bit byte offsets from the logits/H base (max 256MB/64MB, fits).
    unsigned goffL[4], goffH[4];
    unsigned ldsL[2][4], ldsH[2][4];
    #pragma unroll
    for (int j = 0; j < 4; ++j) {
        const int idx = lane + 32 * j;
        const int r   = idx >> 3;
        const int s   = idx & 7;
        goffL[j] = (unsigned)((row0 + r) * C + s * 4) * 4u;
        goffH[j] = (unsigned)(sTgt[r]    * C + s * 4) * 4u;
        #pragma unroll
        for (int b = 0; b < 2; ++b) {
            ldsL[b][j] = (unsigned)(uintptr_t)&sLog[wave][b][r * LSTRIDE + s * 4];
            ldsH[b][j] = (unsigned)(uintptr_t)&sHm [wave][b][r * LSTRIDE + s * 4];
        }
    }

    // issue one chunk's 8 async B128 loads (memory pipe writes LDS directly)
    auto issue_chunk = [&](int ch, int b) {
        const unsigned kb = (unsigned)ch * (CHUNK * 4u);
        #pragma unroll
        for (int j = 0; j < 4; ++j) {
            const unsigned offL = goffL[j] + kb;
            const unsigned offH = goffH[j] + kb;
            asm volatile("global_load_async_to_lds_b128 %0, %1, %2"
                         :: "v"(ldsL[b][j]), "v"(offL), "s"(logits) : "memory");
            asm volatile("global_load_async_to_lds_b128 %0, %1, %2"
                         :: "v"(ldsH[b][j]), "v"(offH), "s"(H) : "memory");
            // warm L2 one more chunk ahead
            __builtin_prefetch((const char*)logits + offL + NWAVES * CHUNK * 4, 0, 3);
            __builtin_prefetch((const char*)H      + offH + NWAVES * CHUNK * 4, 0, 3);
        }
    };

    // ---- phase 2: double-buffered async staging + exp + WMMA, no WG barriers ----
    v8f  acc = {0.f, 0.f, 0.f, 0.f, 0.f, 0.f, 0.f, 0.f};
    float z  = 0.0f;
    const int nchunks  = C / CHUNK;            // 128
    const int mychunks = nchunks / NWAVES;     // 32 (uniform per wave)

    issue_chunk(wave, 0);
    for (int c = 0; c < mychunks; ++c) {
        const int b = c & 1;
        if (c + 1 < mychunks) {
            // WAR: prior DS reads of buffer b^1 must be complete before refill
            asm volatile("s_wait_dscnt 0x0" ::: "memory");
            issue_chunk(wave + (c + 1) * NWAVES, b ^ 1);
            // async loads complete in order: <=8 outstanding => chunk c landed
            asm volatile("s_wait_asynccnt 0x8" ::: "memory");
        } else {
            asm volatile("s_wait_asynccnt 0x0" ::: "memory");
        }

        #pragma unroll
        for (int kk = 0; kk < CHUNK / 4; ++kk) {
            float2 af = *(const float2*)&sLog[wave][b][aoff + 4 * kk];
            float2 bf = *(const float2*)&sHm [wave][b][aoff + 4 * kk];
            float e0 = __expf(af.x - mrow);
            float e1 = __expf(af.y - mrow);
            z += e0 + e1;
            v2f a = {e0, e1};
            v2f bb = {bf.x, bf.y};
            // D = A(16x4) x B(4x16) + C ; diag(m) = sum_k exp(x[m][k]-M[m])*H[t_m][k]
            acc = __builtin_amdgcn_wmma_f32_16x16x4_f32(
                      false, a, false, bb, (short)0, acc, false, false);
        }
    }

    // ---- combine partition function Z per row ----
    z += __shfl_xor(z, 16, 32);
    if (lane < 16) atomicAdd(&sZ[fi], z);

    // ---- extract diagonal of accumulator, sum across waves ----
    // C/D layout: VGPR r: lanes 0-15 -> (M=r, N=lane); lanes 16-31 -> (M=8+r, N=lane-16)
    #pragma unroll
    for (int r = 0; r < 8; ++r) {
        float v = acc[r];
        if (lane == r)      atomicAdd(&sDot[r],     v);
        if (lane == 24 + r) atomicAdd(&sDot[r + 8], v);
    }
    __syncthreads();

    // ---- per-row loss, reduce 16 rows, write per-WG partial ----
    if (wave == 0) {
        const int m  = lane & 15;
        float Zr  = sZ[m];
        float ce  = -(sXt[m] - sM[m] - __logf(Zr));
        float pen = 1.0f + sDot[m] / Zr;
        float li  = (1.0f - ALPHA_) * ce + ALPHA_ * pen;
        float v   = (lane < 16) ? li : 0.0f;
        #pragma unroll
        for (int off = 8; off; off >>= 1) v += __shfl_xor(v, off, 32);
        if (lane == 0) partial[blockIdx.x] = v / (float)Bn;
    }
}

// Deterministic final reduction (single wave, fixed order every call).
__global__ void hier_loss_reduce(const float* __restrict__ partial,
                                 float* __restrict__ out, int n)
{
    float s = 0.0f;
    for (int i = threadIdx.x; i < n; i += 32) s += partial[i];
    #pragma unroll
    for (int off = 16; off; off >>= 1) s += __shfl_xor(s, off, 32);
    if (threadIdx.x == 0) out[0] = s;
}

extern "C" void kernel_launch(void* const* d_in, const int* in_sizes, int n_in,
                              void* d_out, int out_size, void* d_ws, size_t ws_size,
                              hipStream_t stream)
{
    const float* logits  = (const float*)d_in[0];
    const int*   targets = (const int*)d_in[1];
    const float* H       = (const float*)d_in[2];

    const int Bn   = in_sizes[1];          // 16384
    const int C    = in_sizes[0] / Bn;     // 4096
    const int nblk = Bn / ROWS;            // 1024

    float* partial = (float*)d_ws;         // nblk floats of scratch

    hier_loss_kernel<<<nblk, BLOCK, 0, stream>>>(logits, targets, H, partial, Bn, C);
    hier_loss_reduce<<<1, 32, 0, stream>>>(partial, (float*)d_out, nblk);
}